// GroupQueryAttention_33234456936579
// MI455X (gfx1250) — compile-verified
//
#include <hip/hip_runtime.h>

typedef float v2f __attribute__((ext_vector_type(2)));
typedef float v8f __attribute__((ext_vector_type(8)));

#define Bn   4
#define Sn   1024
#define Hn   32
#define HKV  8
#define Dn   64
#define EMB  2048
#define KVW  512          // NUM_KV_HEADS * HEAD_DIM
#define NKT  (Sn / 16)    // 64 key tiles

__device__ __forceinline__ v8f wmma_f32(v2f a, v2f b, v8f c) {
    // D = A(16x4) * B(4x16) + C(16x16), all f32
    return __builtin_amdgcn_wmma_f32_16x16x4_f32(
        /*neg_a=*/false, a, /*neg_b=*/false, b,
        /*c_mod=*/(short)0, c, /*reuse_a=*/false, /*reuse_b=*/false);
}

__global__ __launch_bounds__(32)
void gqa_fwd_kernel(const float* __restrict__ Q, const float* __restrict__ K,
                    const float* __restrict__ V, float* __restrict__ Out,
                    float* __restrict__ Attn) {
    __shared__ __align__(16) float lds_p[16 * 16];   // p tile (C-layout -> A-layout + wide store)
    __shared__ __align__(16) float lds_v[16 * 64];   // staged V tile / out writeback bounce

    const int lane = threadIdx.x;      // 0..31
    const int lo   = lane & 15;
    const int hi   = lane >> 4;        // 0 or 1

    const int bid = blockIdx.x;
    const int qt  = bid & (NKT - 1);          // q tile index
    const int h   = (bid >> 6) & (Hn - 1);    // head
    const int b   = bid >> 11;                // batch
    const int hkv = h >> 2;                   // kv head (Q_PER_KV = 4)

    // ---- load Q tile as 16 A-fragments (K chunks of 4), pre-scaled by 1/sqrt(d)
    const float* qbase = Q + ((size_t)b * Sn + (size_t)qt * 16 + lo) * EMB + h * Dn;
    v2f qa[16];
#pragma unroll
    for (int i = 0; i < 16; ++i) {
        v2f t = *(const v2f*)(qbase + 4 * i + 2 * hi);
        t.x *= 0.125f;
        t.y *= 0.125f;
        qa[i] = t;
    }

    // ---- pass A: per-LANE online softmax stats (no cross-lane work in the loop).
    // Within a 16-lane group, lane `lo` owns column (kt*16 + lo) of row M = v + 8*hi,
    // so per-lane (max, sum) streams are independent; combine across lanes once at end.
    float lmax[8], lsum[8];
#pragma unroll
    for (int v = 0; v < 8; ++v) { lmax[v] = -1e30f; lsum[v] = 0.0f; }

    for (int kt = 0; kt < NKT; ++kt) {
        const float* kbase = K + ((size_t)b * Sn + (size_t)kt * 16 + lo) * KVW + hkv * Dn;
        v8f c = {};
#pragma unroll
        for (int i = 0; i < 16; ++i) {
            v2f kb = *(const v2f*)(kbase + 4 * i + 2 * hi);
            c = wmma_f32(qa[i], kb, c);
        }
        // single-exp online update: one of exp(lmax-mnew)/exp(s-mnew) is exp(0)
#pragma unroll
        for (int v = 0; v < 8; ++v) {
            float s = c[v];
            float d = s - lmax[v];
            float e = __expf(-fabsf(d));
            bool gt = d > 0.0f;
            lsum[v] = gt ? (lsum[v] * e + 1.0f) : (lsum[v] + e);
            lmax[v] = gt ? s : lmax[v];
        }
    }

    // ---- combine per-lane stats across the 16-lane group (once)
    float rmax[8], rinv[8];
#pragma unroll
    for (int v = 0; v < 8; ++v) {
        float m = lmax[v];
        m = fmaxf(m, __shfl_xor(m, 1, 16));
        m = fmaxf(m, __shfl_xor(m, 2, 16));
        m = fmaxf(m, __shfl_xor(m, 4, 16));
        m = fmaxf(m, __shfl_xor(m, 8, 16));
        float partial = lsum[v] * __expf(lmax[v] - m);
        partial += __shfl_xor(partial, 1, 16);
        partial += __shfl_xor(partial, 2, 16);
        partial += __shfl_xor(partial, 4, 16);
        partial += __shfl_xor(partial, 8, 16);
        rmax[v] = m;
        rinv[v] = 1.0f / partial;
    }

    // ---- pass B: recompute scores, normalize, store attn (wide), out += p @ V
    v8f oc0 = {}, oc1 = {}, oc2 = {}, oc3 = {};

    for (int kt = 0; kt < NKT; ++kt) {
        const float* kbase = K + ((size_t)b * Sn + (size_t)kt * 16 + lo) * KVW + hkv * Dn;
        v8f c = {};
#pragma unroll
        for (int i = 0; i < 16; ++i) {
            v2f kb = *(const v2f*)(kbase + 4 * i + 2 * hi);
            c = wmma_f32(qa[i], kb, c);
        }

        // stage V tile (16 rows x 64 cols) into LDS: 256 float4, 8 per lane
        // (issued early so the global loads overlap the exp/normalize VALU work)
#pragma unroll
        for (int j = 0; j < 8; ++j) {
            int idx = lane + 32 * j;          // float4 index 0..255
            int row = idx >> 4;
            int c4  = idx & 15;
            const float4* src = (const float4*)(
                V + ((size_t)b * Sn + (size_t)kt * 16 + row) * KVW + hkv * Dn + c4 * 4);
            *(float4*)&lds_v[row * 64 + c4 * 4] = *src;
        }

        // normalized probabilities -> LDS (row-major)
#pragma unroll
        for (int v = 0; v < 8; ++v) {
            const int M = v + 8 * hi;
            float p = __expf(c[v] - rmax[v]) * rinv[v];
            lds_p[M * 16 + lo] = p;
        }
        __syncthreads();

        // wide attn store: 64 float4 per tile, 2 per lane -> global_store_b128
        {
            const size_t abase =
                (((size_t)b * Hn + h) * Sn + (size_t)qt * 16) * Sn + (size_t)kt * 16;
#pragma unroll
            for (int t = 0; t < 2; ++t) {
                int idx = lane * 2 + t;       // 0..63
                int row = idx >> 2;
                int ch  = idx & 3;
                *(float4*)(Attn + abase + (size_t)row * Sn + ch * 4) =
                    *(const float4*)&lds_p[row * 16 + ch * 4];
            }
        }

        // out[16x64] += p[16x16] @ Vtile[16x64]  (4 n-tiles x 4 K-chunks)
#pragma unroll
        for (int j = 0; j < 4; ++j) {
            const int kk = 4 * j + 2 * hi;
            v2f pa = *(const v2f*)&lds_p[lo * 16 + kk];
            v2f vb0, vb1, vb2, vb3;
            vb0.x = lds_v[kk * 64 +  0 + lo]; vb0.y = lds_v[(kk + 1) * 64 +  0 + lo];
            vb1.x = lds_v[kk * 64 + 16 + lo]; vb1.y = lds_v[(kk + 1) * 64 + 16 + lo];
            vb2.x = lds_v[kk * 64 + 32 + lo]; vb2.y = lds_v[(kk + 1) * 64 + 32 + lo];
            vb3.x = lds_v[kk * 64 + 48 + lo]; vb3.y = lds_v[(kk + 1) * 64 + 48 + lo];
            oc0 = wmma_f32(pa, vb0, oc0);
            oc1 = wmma_f32(pa, vb1, oc1);
            oc2 = wmma_f32(pa, vb2, oc2);
            oc3 = wmma_f32(pa, vb3, oc3);
        }
        __syncthreads();   // protect lds_p / lds_v reuse next iteration
    }

    // ---- write out tile [16 x 64] via LDS bounce -> global_store_b128
#pragma unroll
    for (int v = 0; v < 8; ++v) {
        const int M = v + 8 * hi;
        lds_v[M * 64 +  0 + lo] = oc0[v];
        lds_v[M * 64 + 16 + lo] = oc1[v];
        lds_v[M * 64 + 32 + lo] = oc2[v];
        lds_v[M * 64 + 48 + lo] = oc3[v];
    }
    __syncthreads();
    {
        const size_t obase = ((size_t)b * Sn + (size_t)qt * 16) * EMB + h * Dn;
#pragma unroll
        for (int j = 0; j < 8; ++j) {
            int idx = lane + 32 * j;          // float4 index 0..255
            int row = idx >> 4;
            int c4  = idx & 15;
            *(float4*)(Out + obase + (size_t)row * EMB + c4 * 4) =
                *(const float4*)&lds_v[row * 64 + c4 * 4];
        }
    }
}

extern "C" void kernel_launch(void* const* d_in, const int* in_sizes, int n_in,
                              void* d_out, int out_size, void* d_ws, size_t ws_size,
                              hipStream_t stream) {
    const float* Q = (const float*)d_in[0];
    const float* K = (const float*)d_in[1];
    const float* V = (const float*)d_in[2];

    float* Out  = (float*)d_out;                               // [B, S, EMB]
    float* Attn = Out + (size_t)Bn * Sn * EMB;                 // [B, H, S, S]

    const int blocks = Bn * Hn * NKT;                          // 4 * 32 * 64 = 8192
    gqa_fwd_kernel<<<blocks, 32, 0, stream>>>(Q, K, V, Out, Attn);
}